// WL_9388798509634
// MI455X (gfx1250) — compile-verified
//
#include <hip/hip_runtime.h>
#include <stdint.h>

// WL graph coloring for gfx1250 (MI455X). Irregular integer workload:
// CSR gather + per-wave LDS rank-hash + bucketed 64-bit key sort for
// jnp.unique-style relabeling. Uses CDNA5 async global->LDS DMA
// (global_load_async_to_lds_b32 / s_wait_asynccnt) for neighbor staging.

typedef unsigned int       u32;
typedef unsigned long long u64;

#define WL_M1 2654435761u
#define WL_M2 2246822519u
#define WL_M3 3266489917u
#define WL_M4 668265263u

#define WL_IT     5
#define WL_NB     (1 << 18)     // buckets for key sort (top 18 bits)
#define WL_BSHIFT 46            // 64 - 18
#define WL_MAXD   128           // LDS fast-path degree cap (avg degree 32)

// ---------------------------------------------------------------- utilities
__global__ void wl_zero(u32* p, int n) {
  int i = blockIdx.x * 256 + threadIdx.x;
  if (i < n) p[i] = 0u;
}

__global__ void wl_copy(u32* d, const u32* s, int n) {
  int i = blockIdx.x * 256 + threadIdx.x;
  if (i < n) d[i] = s[i];
}

// ------------------------------------------------------- hierarchical scan
// exclusive scan, n <= 256*1024
__global__ void wl_scan_block(const u32* in, u32* out, u32* bsums, int n) {
  __shared__ u32 tmp[256];
  int tid = threadIdx.x;
  int i = blockIdx.x * 256 + tid;
  u32 v = (i < n) ? in[i] : 0u;
  tmp[tid] = v;
  __syncthreads();
  for (int off = 1; off < 256; off <<= 1) {
    u32 add = (tid >= off) ? tmp[tid - off] : 0u;
    __syncthreads();
    tmp[tid] += add;
    __syncthreads();
  }
  if (i < n) out[i] = tmp[tid] - v;            // exclusive
  if (tid == 255) bsums[blockIdx.x] = tmp[255];
}

__global__ void wl_scan_top(u32* b, int nb) {   // single block, 1024 threads
  __shared__ u32 tmp[1024];
  int tid = threadIdx.x;
  u32 v = (tid < nb) ? b[tid] : 0u;
  tmp[tid] = v;
  __syncthreads();
  for (int off = 1; off < 1024; off <<= 1) {
    u32 add = (tid >= off) ? tmp[tid - off] : 0u;
    __syncthreads();
    tmp[tid] += add;
    __syncthreads();
  }
  if (tid < nb) b[tid] = tmp[tid] - v;          // exclusive
}

__global__ void wl_scan_add(u32* out, const u32* b, int n) {
  int i = blockIdx.x * 256 + threadIdx.x;
  if (i < n) out[i] += b[blockIdx.x];
}

// ------------------------------------------------------------- CSR build
__global__ void wl_count_rows(const int* row, u32* deg, int e) {
  int i = blockIdx.x * 256 + threadIdx.x;
  if (i < e) atomicAdd(&deg[row[i]], 1u);
}

__global__ void wl_csr_scatter(const int* row, const int* col, u32* cur,
                               int* col_csr, int e) {
  int i = blockIdx.x * 256 + threadIdx.x;
  if (i >= e) return;
  u32 p = atomicAdd(&cur[row[i]], 1u);
  col_csr[p] = col[i];
}

// ------------------------------------------------- per-node hash (1 wave32)
__global__ void wl_hash(const u32* rowp, const int* col_csr, const int* colors,
                        u64* keys, int n, int e) {
  __shared__ u32 buf[8][WL_MAXD];               // 8 waves per 256-thr block
  int wave = threadIdx.x >> 5;
  int lane = threadIdx.x & 31;
  int node = blockIdx.x * 8 + wave;
  if (node >= n) return;

  u32 start = rowp[node];
  u32 end   = (node + 1 < n) ? rowp[node + 1] : (u32)e;
  int d = (int)(end - start);
  u32 s1 = 0u, s2 = 0u;

  if (d <= WL_MAXD) {
    // Stage neighbor colors into LDS with the CDNA5 async DMA path:
    // per-lane gather address -> LDS, tracked by ASYNCcnt.
    for (int i = lane; i < d; i += 32) {
      int c = col_csr[start + (u32)i];
      u64 gaddr = (u64)(size_t)(colors + c);
      u32 loff  = (u32)(size_t)&buf[wave][i];   // generic->LDS offset (low 32b)
      asm volatile("global_load_async_to_lds_b32 %0, %1, off"
                   :: "v"(loff), "v"(gaddr) : "memory");
    }
    asm volatile("s_wait_asynccnt 0" ::: "memory");
    // tie-aware rank (== sorted position; sum is invariant to tie order)
    for (int i = lane; i < d; i += 32) {
      u32 ci = buf[wave][i];
      u32 pos = 0u;
      for (int j = 0; j < d; ++j) {
        u32 cj = buf[wave][j];
        pos += (u32)((cj < ci) || (cj == ci && j < i));
      }
      s1 += (ci + 1u) * (pos * WL_M1 + WL_M2);
      s2 += (ci + 1u) * (pos * WL_M3 + WL_M4);
    }
  } else {
    // rare fat-node fallback straight from L2
    for (int i = lane; i < d; i += 32) {
      u32 ci = (u32)colors[col_csr[start + (u32)i]];
      u32 pos = 0u;
      for (int j = 0; j < d; ++j) {
        u32 cj = (u32)colors[col_csr[start + (u32)j]];
        pos += (u32)((cj < ci) || (cj == ci && j < i));
      }
      s1 += (ci + 1u) * (pos * WL_M1 + WL_M2);
      s2 += (ci + 1u) * (pos * WL_M3 + WL_M4);
    }
  }

  // wave32 reduction (exact uint32 wraparound adds)
  for (int off = 16; off > 0; off >>= 1) {
    s1 += (u32)__shfl_xor((unsigned int)s1, off, 32);
    s2 += (u32)__shfl_xor((unsigned int)s2, off, 32);
  }
  if (lane == 0) {
    u32 own = (u32)colors[node] + 1u;
    u32 k0 = s1 * WL_M2 + own * WL_M1;          // key2[:,0] (lex-major)
    u32 k1 = s2 * WL_M4 + own * WL_M3;          // key2[:,1]
    keys[node] = ((u64)k0 << 32) | (u64)k1;
  }
}

// --------------------------------------------- bucketed sort + rank relabel
__global__ void wl_bucket_count(const u64* keys, u32* bc, int n) {
  int i = blockIdx.x * 256 + threadIdx.x;
  if (i < n) atomicAdd(&bc[(u32)(keys[i] >> WL_BSHIFT)], 1u);
}

__global__ void wl_bucket_scatter(const u64* keys, u32* cur, u64* skey,
                                  int* sidx, int n) {
  int i = blockIdx.x * 256 + threadIdx.x;
  if (i >= n) return;
  u64 k = keys[i];
  u32 p = atomicAdd(&cur[(u32)(k >> WL_BSHIFT)], 1u);
  skey[p] = k;
  sidx[p] = i;
}

__global__ void wl_bucket_sort(const u32* bptr, const u32* bc, u64* skey,
                               int* sidx, int nb) {
  int b = blockIdx.x * 256 + threadIdx.x;
  if (b >= nb) return;
  int s = (int)bptr[b];
  int m = (int)bc[b];                            // avg 0.76 keys per bucket
  for (int i = 1; i < m; ++i) {
    u64 k = skey[s + i];
    int id = sidx[s + i];
    int j = i - 1;
    while (j >= 0 && skey[s + j] > k) {          // strict >: equal keys O(1)
      skey[s + j + 1] = skey[s + j];
      sidx[s + j + 1] = sidx[s + j];
      --j;
    }
    skey[s + j + 1] = k;
    sidx[s + j + 1] = id;
  }
}

__global__ void wl_flags(const u64* skey, u32* fl, int n) {
  int i = blockIdx.x * 256 + threadIdx.x;
  if (i >= n) return;
  fl[i] = (i == 0 || skey[i] != skey[i - 1]) ? 1u : 0u;
}

__global__ void wl_write_colors(const u64* skey, const int* sidx,
                                const u32* rex, int* cn, int n) {
  int i = blockIdx.x * 256 + threadIdx.x;
  if (i >= n) return;
  u32 f = (i == 0 || skey[i] != skey[i - 1]) ? 1u : 0u;
  cn[sidx[i]] = (int)(rex[i] + f - 1u);          // rank among sorted distinct
}

__global__ void wl_to_float(const int* c, float* o, int n) {
  int i = blockIdx.x * 256 + threadIdx.x;
  if (i < n) o[i] = (float)c[i];
}

// ------------------------------------------------------------------- driver
extern "C" void kernel_launch(void* const* d_in, const int* in_sizes, int n_in,
                              void* d_out, int out_size, void* d_ws, size_t ws_size,
                              hipStream_t stream) {
  const int* x  = (const int*)d_in[0];
  const int* ei = (const int*)d_in[1];
  const int N = in_sizes[0];
  const int E = in_sizes[1] / 2;
  const int* row = ei;
  const int* col = ei + E;
  (void)n_in; (void)ws_size;

  char* base = (char*)d_ws;
  size_t off = 0;
  auto alloc = [&](size_t bytes) -> char* {
    char* p = base + off;
    off = (off + bytes + 255) & ~(size_t)255;
    return p;
  };
  u32* deg    = (u32*)alloc((size_t)N * 4);
  u32* rowp   = (u32*)alloc((size_t)N * 4);
  u32* cur    = (u32*)alloc((size_t)WL_NB * 4);   // CSR cursors, then bucket cursors
  int* colcsr = (int*)alloc((size_t)E * 4);
  int* colA   = (int*)alloc((size_t)N * 4);
  int* colB   = (int*)alloc((size_t)N * 4);
  u64* keys   = (u64*)alloc((size_t)N * 8);
  u64* skey   = (u64*)alloc((size_t)N * 8);
  int* sidx   = (int*)alloc((size_t)N * 4);
  u32* bc     = (u32*)alloc((size_t)WL_NB * 4);
  u32* bp     = (u32*)alloc((size_t)WL_NB * 4);
  u32* flags  = (u32*)alloc((size_t)N * 4);
  u32* rex    = (u32*)alloc((size_t)N * 4);
  u32* bsums  = (u32*)alloc((size_t)1024 * 4);

  auto gr = [](int n) { return (n + 255) / 256; };
  auto scan = [&](const u32* in, u32* out, int n) {   // n <= 256*1024
    int nb = gr(n);
    wl_scan_block<<<nb, 256, 0, stream>>>(in, out, bsums, n);
    wl_scan_top<<<1, 1024, 0, stream>>>(bsums, nb);
    wl_scan_add<<<nb, 256, 0, stream>>>(out, bsums, n);
  };

  // ---- CSR (edge_index is loop-invariant: build once per launch)
  wl_zero<<<gr(N), 256, 0, stream>>>(deg, N);
  wl_count_rows<<<gr(E), 256, 0, stream>>>(row, deg, E);
  scan(deg, rowp, N);
  wl_copy<<<gr(N), 256, 0, stream>>>(cur, rowp, N);
  wl_csr_scatter<<<gr(E), 256, 0, stream>>>(row, col, cur, colcsr, E);

  // ---- colors init
  wl_copy<<<gr(N), 256, 0, stream>>>((u32*)colA, (const u32*)x, N);

  int* ccur = colA;
  int* cnext = colB;
  for (int it = 0; it < WL_IT; ++it) {
    wl_hash<<<(N + 7) / 8, 256, 0, stream>>>(rowp, colcsr, ccur, keys, N, E);

    wl_zero<<<gr(WL_NB), 256, 0, stream>>>(bc, WL_NB);
    wl_bucket_count<<<gr(N), 256, 0, stream>>>(keys, bc, N);
    scan(bc, bp, WL_NB);
    wl_copy<<<gr(WL_NB), 256, 0, stream>>>(cur, bp, WL_NB);
    wl_bucket_scatter<<<gr(N), 256, 0, stream>>>(keys, cur, skey, sidx, N);
    wl_bucket_sort<<<gr(WL_NB), 256, 0, stream>>>(bp, bc, skey, sidx, WL_NB);

    wl_flags<<<gr(N), 256, 0, stream>>>(skey, flags, N);
    scan(flags, rex, N);
    wl_write_colors<<<gr(N), 256, 0, stream>>>(skey, sidx, rex, cnext, N);

    int* t = ccur; ccur = cnext; cnext = t;
  }

  wl_to_float<<<gr(N), 256, 0, stream>>>(ccur, (float*)d_out, N);
  (void)out_size;
}